// GraphTransformer_55336358642853
// MI455X (gfx1250) — compile-verified
//
#include <hip/hip_runtime.h>
#include <hip/hip_bf16.h>
#include <math.h>

// ---------------------------------------------------------------------------
// CDNA5 WMMA (wave32): V_WMMA_F32_16X16X4_F32
//   A: 16x4 f32  -> 2 VGPRs/lane (v2f)   lane = row, half selects K pair
//   B:  4x16 f32 -> 2 VGPRs/lane (v2f)   lane = col, half selects K pair
//   C/D: 16x16 f32 -> 8 VGPRs/lane (v8f) VGPR r -> row r + 8*half
// ---------------------------------------------------------------------------
typedef __attribute__((ext_vector_type(2))) float v2f;
typedef __attribute__((ext_vector_type(8))) float v8f;

#define GT_N        50000     // multiple of 16
#define GT_E        400000    // multiple of 16
#define GT_NCLS     500
#define GT_EPS      1e-5f

__device__ __forceinline__ float gelu_exact(float x) {
    return 0.5f * x * (1.0f + erff(x * 0.70710678118654752f));
}

__device__ __forceinline__ void atomicMaxF(float* addr, float val) {
    int* ai = (int*)addr;
    int old = __float_as_int(*addr);
    while (__int_as_float(old) < val) {
        int assumed = old;
        old = atomicCAS(ai, assumed, __float_as_int(val));
        if (old == assumed) break;
    }
}

// -------------------------------- utility ----------------------------------
__global__ void fill_f32(float* __restrict__ p, float v, int n) {
    int i = blockIdx.x * blockDim.x + threadIdx.x;
    if (i < n) p[i] = v;
}

__global__ void zero2(float* p) { p[0] = 0.f; p[1] = 0.f; }

// out[i*64+c] = table[idx[i]*64+c]
__global__ void gather_rows64(const int* __restrict__ idx,
                              const float* __restrict__ table,
                              float* __restrict__ out, int total) {
    int i = blockIdx.x * blockDim.x + threadIdx.x;
    if (i >= total) return;
    int row = idx[i >> 6];
    out[i] = table[(size_t)row * 64 + (i & 63)];
}

// ------------------------------ graph layernorm -----------------------------
__global__ void ln_reduce(const float* __restrict__ x, int n, float* __restrict__ sums) {
    __shared__ float ssum[256];
    __shared__ float ssq[256];
    float s = 0.f, q = 0.f;
    for (int i = blockIdx.x * blockDim.x + threadIdx.x; i < n; i += gridDim.x * blockDim.x) {
        float v = x[i];
        s += v; q += v * v;
    }
    ssum[threadIdx.x] = s; ssq[threadIdx.x] = q;
    __syncthreads();
    for (int off = 128; off > 0; off >>= 1) {
        if ((int)threadIdx.x < off) {
            ssum[threadIdx.x] += ssum[threadIdx.x + off];
            ssq[threadIdx.x]  += ssq[threadIdx.x + off];
        }
        __syncthreads();
    }
    if (threadIdx.x == 0) {
        atomicAdd(&sums[0], ssum[0]);
        atomicAdd(&sums[1], ssq[0]);
    }
}

__global__ void ln_apply(const float* __restrict__ x,
                         const float* __restrict__ w, const float* __restrict__ b,
                         const float* __restrict__ sums, float inv_n,
                         float* __restrict__ y, int total) {
    int i = blockIdx.x * blockDim.x + threadIdx.x;
    if (i >= total) return;
    float mean = sums[0] * inv_n;
    float var  = sums[1] * inv_n - mean * mean;
    float sd   = sqrtf(fmaxf(var, 0.f));
    int c = i & 63;
    y[i] = (x[i] - mean) / (sd + GT_EPS) * w[c] + b[c];
}

// ------------------------------ WMMA f32 GEMM -------------------------------
// Y[M,Nc] = act(X[M,K] @ W[K,Nc] + bias) + addend
// Requirements: M % 16 == 0 (true for N=50000, E=400000), Nc % 16 == 0.
// Block = 256 threads = 8 waves; all waves share one 16-column W panel staged
// in LDS (transposed, +2 dword pad stride -> conflict-free ds_load_b64).
// Wave w computes the 16x16 tile (blockIdx.x*8+w, blockIdx.y).
template<int K>
__global__ void wmma_gemm_f32(const float* __restrict__ X,
                              const float* __restrict__ W,
                              const float* __restrict__ bias,
                              const float* __restrict__ addend,
                              float* __restrict__ Y,
                              int M, int Nc, int act) {
    constexpr int SK = K + 2;                 // padded LDS stride (banks)
    __shared__ float Wt[16 * SK];             // W panel, transposed [col][k]

    const int tid  = threadIdx.x;
    const int lane = tid & 31;
    const int wave = tid >> 5;
    const int cn   = blockIdx.y;              // column-tile index
    const int ncol_base = cn << 4;

    // stage W panel: Wt[c][kk] = W[kk*Nc + ncol_base + c]
    #pragma unroll
    for (int i = tid; i < 16 * K; i += 256) {
        int c  = i / K;
        int kk = i - c * K;
        Wt[c * SK + kk] = W[kk * Nc + ncol_base + c];
    }
    __syncthreads();

    const int tiles_m = M >> 4;
    const int tm = blockIdx.x * 8 + wave;
    if (tm >= tiles_m) return;                // wave-uniform, after barrier

    const int hi = lane >> 4;                 // lane half: K-pair select
    const int lo = lane & 15;
    const int mrow = (tm << 4) + lo;          // A row for this lane
    const int ncol = ncol_base + lo;          // B/D column for this lane

    // preload full per-lane A row: pairs {k0+2hi, k0+2hi+1}, k0 += 4
    v2f areg[K / 4];
    const float* __restrict__ Xr = X + (unsigned)mrow * K + 2 * hi;
    #pragma unroll
    for (int j = 0; j < K / 4; ++j)
        areg[j] = *(const v2f*)(Xr + 4 * j);

    // WMMA chain fed from registers (A) and LDS (B)
    v8f acc = {};
    const float* __restrict__ Wrow = &Wt[lo * SK + 2 * hi];
    #pragma unroll
    for (int j = 0; j < K / 4; ++j) {
        v2f b = *(const v2f*)(Wrow + 4 * j);
        acc = __builtin_amdgcn_wmma_f32_16x16x4_f32(false, areg[j], false, b,
                                                    (short)0, acc, false, false);
    }

    const float bcol = bias ? bias[ncol] : 0.f;
    const int rbase = (tm << 4) + 8 * hi;
    #pragma unroll
    for (int r = 0; r < 8; ++r) {
        const unsigned row = rbase + r;       // D layout: VGPR r -> row r + 8*half
        float val = acc[r] + bcol;
        if (act == 1) val = gelu_exact(val);
        if (addend) val += addend[row * (unsigned)Nc + ncol];
        Y[row * (unsigned)Nc + ncol] = val;
    }
}

// ------------------------------- attention ----------------------------------
__global__ void attn_alpha(const float* __restrict__ q, const float* __restrict__ k,
                           const float* __restrict__ ep,
                           const int* __restrict__ src, const int* __restrict__ dst,
                           float* __restrict__ alpha, float* __restrict__ mx, int E) {
    int idx = blockIdx.x * blockDim.x + threadIdx.x;
    if (idx >= E * 2) return;
    int e = idx >> 1, hd = idx & 1;
    int s = src[e], d = dst[e];
    const float4* qp  = (const float4*)(q  + (size_t)d * 128 + hd * 64);
    const float4* kp  = (const float4*)(k  + (size_t)s * 128 + hd * 64);
    const float4* epp = (const float4*)(ep + (size_t)e * 128 + hd * 64);
    float acc = 0.f;
#pragma unroll
    for (int i = 0; i < 16; ++i) {
        float4 a = qp[i], b = kp[i], c = epp[i];
        acc += a.x * (b.x + c.x) + a.y * (b.y + c.y)
             + a.z * (b.z + c.z) + a.w * (b.w + c.w);
    }
    float al = acc * 0.125f;                  // 1/sqrt(64)
    alpha[idx] = al;
    atomicMaxF(&mx[d * 2 + hd], al);
}

__global__ void attn_exp(const float* __restrict__ alpha, const float* __restrict__ mx,
                         const int* __restrict__ dst,
                         float* __restrict__ ealpha, float* __restrict__ ssum, int E) {
    int idx = blockIdx.x * blockDim.x + threadIdx.x;
    if (idx >= E * 2) return;
    int e = idx >> 1, hd = idx & 1;
    int d = dst[e];
    float ea = expf(alpha[idx] - mx[d * 2 + hd]);
    ealpha[idx] = ea;
    atomicAdd(&ssum[d * 2 + hd], ea);
}

__global__ void attn_msg(const float* __restrict__ v, const float* __restrict__ ep,
                         const float* __restrict__ ealpha, const float* __restrict__ ssum,
                         const int* __restrict__ src, const int* __restrict__ dst,
                         float* __restrict__ agg, int E) {
    int idx = blockIdx.x * blockDim.x + threadIdx.x;
    if (idx >= E * 128) return;
    int e = idx >> 7, r = idx & 127, hd = r >> 6;
    int s = src[e], d = dst[e];
    float coef = ealpha[e * 2 + hd] / (ssum[d * 2 + hd] + 1e-16f);
    float val = (v[(size_t)s * 128 + r] + ep[(size_t)e * 128 + r]) * coef;
    atomicAdd(&agg[(size_t)d * 128 + r], val);
}

__global__ void mean_heads(const float* __restrict__ agg, const float* __restrict__ res,
                           float* __restrict__ out, int total) {
    int i = blockIdx.x * blockDim.x + threadIdx.x;
    if (i >= total) return;
    int n = i >> 6, c = i & 63;
    out[i] = 0.5f * (agg[(size_t)n * 128 + c] + agg[(size_t)n * 128 + 64 + c]) + res[i];
}

__global__ void gelu_inplace(float* __restrict__ p, int n) {
    int i = blockIdx.x * blockDim.x + threadIdx.x;
    if (i < n) p[i] = gelu_exact(p[i]);
}

// --------------------------------- outputs ----------------------------------
__global__ void copy_f32(const float* __restrict__ a, float* __restrict__ b, int n) {
    int i = blockIdx.x * blockDim.x + threadIdx.x;
    if (i < n) b[i] = a[i];
}

__global__ void cls_fill(const float* __restrict__ h, float* __restrict__ outc, int n) {
    int i = blockIdx.x * blockDim.x + threadIdx.x;
    if (i < n) outc[i] = h[i & 63];
}

__global__ void cls_gather(const int* __restrict__ x, const float* __restrict__ h,
                           float* __restrict__ outc, int N, int ncls) {
    int i = blockIdx.x * blockDim.x + threadIdx.x;
    if (i >= N || x[i] != 0) return;
    int rank = 0;
    for (int j = 0; j < i; ++j) rank += (x[j] == 0);
    if (rank < ncls) {
        for (int c = 0; c < 64; ++c)
            outc[(size_t)rank * 64 + c] = h[(size_t)i * 64 + c];
    }
}

// ---------------------------------------------------------------------------
static inline int cdiv(long long a, int b) { return (int)((a + b - 1) / b); }

static void launch_gemm(const float* X, const float* W, const float* bias,
                        const float* addend, float* Y,
                        int M, int K, int Nc, int act, hipStream_t stream) {
    const int tiles_m = M / 16;
    dim3 grid((tiles_m + 7) / 8, Nc / 16);
    if (K == 64)
        wmma_gemm_f32<64><<<grid, 256, 0, stream>>>(X, W, bias, addend, Y, M, Nc, act);
    else
        wmma_gemm_f32<128><<<grid, 256, 0, stream>>>(X, W, bias, addend, Y, M, Nc, act);
}

extern "C" void kernel_launch(void* const* d_in, const int* in_sizes, int n_in,
                              void* d_out, int out_size, void* d_ws, size_t ws_size,
                              hipStream_t stream) {
    (void)in_sizes; (void)n_in; (void)out_size; (void)ws_size;
    const int N = GT_N, E = GT_E;

    const int*   x        = (const int*)d_in[0];
    const int*   src      = (const int*)d_in[1];
    const int*   dst      = src + E;
    const int*   eattr    = (const int*)d_in[3];
    const float* embed    = (const float*)d_in[5];
    const float* embed_ee = (const float*)d_in[6];

    // per-layer dict insertion order:
    // Wq,bq,Wk,bk,Wv,bv,We,ln1_w,ln1_b,ln2_w,ln2_b,Wf1,bf1,Wf2,bf2[,Wp,bp]
    const int lbase[3] = {7, 7 + 17, 7 + 34};

    // ---------------- workspace layout (floats) ----------------
    float* ws = (float*)d_ws;
    size_t o = 0;
    float* h      = ws + o; o += (size_t)N * 64;
    float* hln    = ws + o; o += (size_t)N * 64;
    float* q      = ws + o; o += (size_t)N * 128;
    float* k      = ws + o; o += (size_t)N * 128;
    float* v      = ws + o; o += (size_t)N * 128;
    float* t1     = ws + o; o += (size_t)N * 64;
    float* t2     = ws + o; o += (size_t)N * 64;
    float* agg    = ws + o; o += (size_t)N * 128;
    float* mx     = ws + o; o += (size_t)N * 2;
    float* ssum   = ws + o; o += (size_t)N * 2;
    float* alpha  = ws + o; o += (size_t)E * 2;
    float* ealpha = ws + o; o += (size_t)E * 2;
    float* sums   = ws + o; o += 8;
    float* ee     = ws + o; o += (size_t)E * 64;
    float* eproj  = ws + o; o += (size_t)E * 128;

    const int T = 256;

    // ---------------- embeddings ----------------
    gather_rows64<<<cdiv((long long)N * 64, T), T, 0, stream>>>(x, embed, h, N * 64);
    gather_rows64<<<cdiv((long long)E * 64, T), T, 0, stream>>>(eattr, embed_ee, ee, E * 64);

    const float inv_nd = 1.0f / (float)((long long)N * 64);

    for (int L = 0; L < 3; ++L) {
        const int b = lbase[L];
        const float* Wq    = (const float*)d_in[b + 0];
        const float* bq    = (const float*)d_in[b + 1];
        const float* Wk    = (const float*)d_in[b + 2];
        const float* bk    = (const float*)d_in[b + 3];
        const float* Wv    = (const float*)d_in[b + 4];
        const float* bv    = (const float*)d_in[b + 5];
        const float* We    = (const float*)d_in[b + 6];
        const float* ln1_w = (const float*)d_in[b + 7];
        const float* ln1_b = (const float*)d_in[b + 8];
        const float* ln2_w = (const float*)d_in[b + 9];
        const float* ln2_b = (const float*)d_in[b + 10];
        const float* Wf1   = (const float*)d_in[b + 11];
        const float* bf1   = (const float*)d_in[b + 12];
        const float* Wf2   = (const float*)d_in[b + 13];
        const float* bf2   = (const float*)d_in[b + 14];
        const bool concat  = (L < 2);
        const float* Wp    = concat ? (const float*)d_in[b + 15] : nullptr;
        const float* bp    = concat ? (const float*)d_in[b + 16] : nullptr;

        // LN1 (graph-wide)
        zero2<<<1, 1, 0, stream>>>(sums);
        ln_reduce<<<512, T, 0, stream>>>(h, N * 64, sums);
        ln_apply<<<cdiv((long long)N * 64, T), T, 0, stream>>>(h, ln1_w, ln1_b, sums,
                                                               inv_nd, hln, N * 64);

        // projections (WMMA f32)
        launch_gemm(hln, Wq, bq, nullptr, q, N, 64, 128, 0, stream);
        launch_gemm(hln, Wk, bk, nullptr, k, N, 64, 128, 0, stream);
        launch_gemm(hln, Wv, bv, nullptr, v, N, 64, 128, 0, stream);
        launch_gemm(ee,  We, nullptr, nullptr, eproj, E, 64, 128, 0, stream);

        // attention
        fill_f32<<<cdiv((long long)N * 2, T), T, 0, stream>>>(mx, -INFINITY, N * 2);
        fill_f32<<<cdiv((long long)N * 2, T), T, 0, stream>>>(ssum, 0.f, N * 2);
        fill_f32<<<cdiv((long long)N * 128, T), T, 0, stream>>>(agg, 0.f, N * 128);
        attn_alpha<<<cdiv((long long)E * 2, T), T, 0, stream>>>(q, k, eproj, src, dst,
                                                                alpha, mx, E);
        attn_exp<<<cdiv((long long)E * 2, T), T, 0, stream>>>(alpha, mx, dst,
                                                              ealpha, ssum, E);
        attn_msg<<<cdiv((long long)E * 128, T), T, 0, stream>>>(v, eproj, ealpha, ssum,
                                                                src, dst, agg, E);

        if (concat) {
            launch_gemm(agg, Wp, bp, h, t1, N, 128, 64, 0, stream);
        } else {
            mean_heads<<<cdiv((long long)N * 64, T), T, 0, stream>>>(agg, h, t1, N * 64);
        }

        // LN2 + FFN
        zero2<<<1, 1, 0, stream>>>(sums);
        ln_reduce<<<512, T, 0, stream>>>(t1, N * 64, sums);
        ln_apply<<<cdiv((long long)N * 64, T), T, 0, stream>>>(t1, ln2_w, ln2_b, sums,
                                                               inv_nd, hln, N * 64);
        launch_gemm(hln, Wf1, bf1, nullptr, t2, N, 64, 64, /*gelu*/1, stream);
        launch_gemm(t2,  Wf2, bf2, t1,      h,  N, 64, 64, 0, stream);

        if (L < 2)
            gelu_inplace<<<cdiv((long long)N * 64, T), T, 0, stream>>>(h, N * 64);
    }

    // ---------------- outputs: (h[cls_idx], h) concatenated ----------------
    float* out  = (float*)d_out;
    float* outh = out + (size_t)GT_NCLS * 64;
    cls_fill<<<cdiv((long long)GT_NCLS * 64, T), T, 0, stream>>>(h, out, GT_NCLS * 64);
    cls_gather<<<cdiv(N, T), T, 0, stream>>>(x, h, out, N, GT_NCLS);
    copy_f32<<<cdiv((long long)N * 64, T), T, 0, stream>>>(h, outh, N * 64);
}